// GTClassLocalizationLoss_16973710754130
// MI455X (gfx1250) — compile-verified
//
#include <hip/hip_runtime.h>
#include <stdint.h>

// Problem constants (match reference)
constexpr int Bc = 16, Cc = 80, Hc = 100, Wc = 100, Nc = 32;
constexpr int HW = Hc * Wc;                 // 10000 cells per class slice
constexpr int NBLK = Bc * Nc;               // 512 GT instances
constexpr int TPB = 256;                    // 8 wave32s
constexpr int NT = (HW + TPB - 1) / TPB;    // 40 tiles
constexpr float POS_THR = 0.3f;

__device__ __forceinline__ float giou4(const float4 a, const float4 g) {
    float area_a = (a.z - a.x) * (a.w - a.y);
    float area_g = (g.z - g.x) * (g.w - g.y);
    float ltx = fmaxf(a.x, g.x), lty = fmaxf(a.y, g.y);
    float rbx = fminf(a.z, g.z), rby = fminf(a.w, g.w);
    float iw = fmaxf(rbx - ltx, 0.0f), ih = fmaxf(rby - lty, 0.0f);
    float inter = iw * ih;
    float uni = area_a + area_g - inter;
    float iou = inter / uni;
    float cx0 = fminf(a.x, g.x), cy0 = fminf(a.y, g.y);
    float cx1 = fmaxf(a.z, g.z), cy1 = fmaxf(a.w, g.w);
    float cw = fmaxf(cx1 - cx0, 0.0f), ch = fmaxf(cy1 - cy0, 0.0f);
    float ac = cw * ch;
    return iou - (ac - uni) / ac;
}

// CDNA5 async global->LDS copy: each lane moves 16B. GVS mode:
//   mem = SGPR64 base + VGPR32 byte-offset. Tracked by ASYNCcnt.
__device__ __forceinline__ void async_ld_b128(uint32_t lds_addr, uint64_t sbase,
                                              uint32_t voff) {
    asm volatile("global_load_async_to_lds_b128 %0, %1, %2"
                 :: "v"(lds_addr), "v"(voff), "s"(sbase)
                 : "memory");
}

__global__ void __launch_bounds__(TPB)
gt_loc_loss_kernel(const float* __restrict__ pred,      // [B,C,H,W,4]
                   const float* __restrict__ gt_boxes,  // [B,N,4]
                   const int*   __restrict__ gt_labels, // [B,N]
                   float* __restrict__ partials)        // [NBLK,3]
{
    __shared__ __align__(16) float tiles[2][TPB * 4];   // 2 x 4KB double buffer
    __shared__ float s_val[TPB];
    __shared__ int   s_idx[TPB];

    const int blk = blockIdx.x;           // 0..511
    const int b   = blk / Nc;
    const int n   = blk % Nc;
    const int tid = threadIdx.x;

    const int label = gt_labels[b * Nc + n];
    const float4 gt = *(const float4*)(gt_boxes + (size_t)(b * Nc + n) * 4);

    const float* grid = pred + ((size_t)b * Cc + (size_t)label) * (size_t)HW * 4;
    const uint64_t gbase = (uint64_t)(uintptr_t)grid;

    const uint32_t lds0 = (uint32_t)(uintptr_t)(&tiles[0][0]);
    const uint32_t lds1 = (uint32_t)(uintptr_t)(&tiles[1][0]);

    // ---- Phase 1: streamed GIoU + argmax over all 10000 cells ------------
    float bestV = -1e30f;
    int   bestI = 0x7fffffff;

    // prologue: issue tile 0
    {
        uint32_t cell = (uint32_t)tid;
        uint32_t voff = (cell < (uint32_t)HW ? cell : (uint32_t)(HW - 1)) * 16u;
        async_ld_b128(lds0 + (uint32_t)tid * 16u, gbase, voff);
    }

    for (int t = 0; t < NT; ++t) {
        if (t + 1 < NT) {
            uint32_t cell = (uint32_t)((t + 1) * TPB + tid);
            uint32_t voff = (cell < (uint32_t)HW ? cell : (uint32_t)(HW - 1)) * 16u;
            uint32_t laddr = (((t + 1) & 1) ? lds1 : lds0) + (uint32_t)tid * 16u;
            async_ld_b128(laddr, gbase, voff);
            asm volatile("s_wait_asynccnt 1" ::: "memory");   // tile t landed
        } else {
            asm volatile("s_wait_asynccnt 0" ::: "memory");
        }
        __syncthreads();   // all waves' pieces of tile t are in LDS

        int cell = t * TPB + tid;
        if (cell < HW) {
            const float4 a =
                *(const float4*)&tiles[t & 1][tid * 4];       // ds_load_b128
            float g = giou4(a, gt);
            if (g > bestV) { bestV = g; bestI = cell; }       // first-max kept
        }
        __syncthreads();   // safe to overwrite this buffer next iteration
    }

    // ---- Block argmax reduction (first-occurrence tie-break) -------------
    s_val[tid] = bestV;
    s_idx[tid] = bestI;
    __syncthreads();
    for (int off = TPB / 2; off > 0; off >>= 1) {
        if (tid < off) {
            float v2 = s_val[tid + off];
            int   i2 = s_idx[tid + off];
            if (v2 > s_val[tid] || (v2 == s_val[tid] && i2 < s_idx[tid])) {
                s_val[tid] = v2;
                s_idx[tid] = i2;
            }
        }
        __syncthreads();
    }

    // ---- Phase 2: <=4x4 window accumulation (thread 0, deterministic) ----
    if (tid == 0) {
        float maxv = s_val[0];
        int   idx  = s_idx[0];
        float l1s = 0.0f, gs = 0.0f, cnt = 0.0f;
        if (maxv > POS_THR) {
            int mi = idx / Wc, mj = idx % Wc;
            int r0 = mi - 2 > 0 ? mi - 2 : 0;
            int r1 = mi + 1 < Hc - 1 ? mi + 1 : Hc - 1;
            int c0 = mj - 2 > 0 ? mj - 2 : 0;
            int c1 = mj + 1 < Wc - 1 ? mj + 1 : Wc - 1;
            for (int r = r0; r <= r1; ++r) {
                for (int c = c0; c <= c1; ++c) {
                    const float4 a = *(const float4*)(grid + (size_t)(r * Wc + c) * 4);
                    float g = giou4(a, gt);
                    l1s += 0.25f * (fabsf(a.x - gt.x) + fabsf(a.y - gt.y) +
                                    fabsf(a.z - gt.z) + fabsf(a.w - gt.w));
                    gs  += 1.0f - g;
                    cnt += 1.0f;
                }
            }
        }
        partials[3 * blk + 0] = l1s;
        partials[3 * blk + 1] = gs;
        partials[3 * blk + 2] = cnt;
    }
}

__global__ void __launch_bounds__(TPB)
finalize_kernel(const float* __restrict__ partials, float* __restrict__ out) {
    __shared__ float sa[TPB], sb[TPB], sc[TPB];
    int tid = threadIdx.x;
    float a = 0.0f, b = 0.0f, c = 0.0f;
    for (int i = tid; i < NBLK; i += TPB) {
        a += partials[3 * i + 0];
        b += partials[3 * i + 1];
        c += partials[3 * i + 2];
    }
    sa[tid] = a; sb[tid] = b; sc[tid] = c;
    __syncthreads();
    for (int off = TPB / 2; off > 0; off >>= 1) {
        if (tid < off) {
            sa[tid] += sa[tid + off];
            sb[tid] += sb[tid + off];
            sc[tid] += sc[tid + off];
        }
        __syncthreads();
    }
    if (tid == 0) {
        float denom = fmaxf(sc[0], 1.0f);
        out[0] = (sa[0] + 2.0f * sb[0]) / denom;   // L1 + 2*GIoU loss
    }
}

extern "C" void kernel_launch(void* const* d_in, const int* in_sizes, int n_in,
                              void* d_out, int out_size, void* d_ws, size_t ws_size,
                              hipStream_t stream) {
    const float* pred   = (const float*)d_in[0];   // [B,C,H,W,4] f32
    const float* gtb    = (const float*)d_in[1];   // [B,N,4] f32
    const int*   gtl    = (const int*)d_in[2];     // [B,N] i32
    float* partials     = (float*)d_ws;            // NBLK*3 floats
    float* out          = (float*)d_out;           // scalar f32

    gt_loc_loss_kernel<<<NBLK, TPB, 0, stream>>>(pred, gtb, gtl, partials);
    finalize_kernel<<<1, TPB, 0, stream>>>(partials, out);
}